// GINConv_2705829396723
// MI455X (gfx1250) — compile-verified
//
#include <hip/hip_runtime.h>

#define N_NODES 10000
#define N_EDGES 640000
#define D_FEAT  128

typedef __attribute__((ext_vector_type(2))) float v2f;
typedef __attribute__((ext_vector_type(8))) float v8f;

// ---------------------------------------------------------------------------
// ws layout (all fp32 unless noted):
//   [0, N*D)              neigh   (scatter-add accumulator)
//   [N*D, N*D+E)          e_raw   (per-edge score)
//   [N*D+E, N*D+2E)       e_exp   (exp(e - max))
//   [N*D+2E, +N) (uint)   mkey    (monotone-encoded segment max)
//   [.., +N)              denom   (segment sum of e_exp)
// total ~10.3 MB
// ---------------------------------------------------------------------------

// Monotone float->uint key: order-preserving, so atomicMax(u32) == float max.
__device__ __forceinline__ unsigned f2key(float f) {
  unsigned b = __float_as_uint(f);
  return (b & 0x80000000u) ? ~b : (b | 0x80000000u);
}
__device__ __forceinline__ float key2f(unsigned k) {
  unsigned b = (k & 0x80000000u) ? (k ^ 0x80000000u) : ~k;
  return __uint_as_float(b);
}

__global__ void __launch_bounds__(256) k_init(float* __restrict__ neigh,
                                              unsigned* __restrict__ mkey,
                                              float* __restrict__ denom) {
  int i = blockIdx.x * 256 + threadIdx.x;
  if (i < N_NODES * D_FEAT) neigh[i] = 0.0f;
  if (i < N_NODES) { mkey[i] = 0u; denom[i] = 0.0f; }
}

// One wave32 per edge: lane handles 4 contiguous floats (float4 = b128 load).
// Scatter-add neigh[dst] += feat[src]; wave-reduce dot(feat[src], tanh(feat[dst])).
__global__ void __launch_bounds__(256) k_edge(const float* __restrict__ feat,
                                              const int* __restrict__ src,
                                              const int* __restrict__ dst,
                                              float* __restrict__ neigh,
                                              float* __restrict__ e_raw,
                                              unsigned* __restrict__ mkey) {
  const int wave = threadIdx.x >> 5;
  const int lane = threadIdx.x & 31;
  const int edge = blockIdx.x * 8 + wave;
  if (edge >= N_EDGES) return;

  const int s = src[edge];
  const int d = dst[edge];

  const float4 fs = ((const float4*)(feat + (size_t)s * D_FEAT))[lane];
  const float4 fd = ((const float4*)(feat + (size_t)d * D_FEAT))[lane];

  float* np = neigh + (size_t)d * D_FEAT + lane * 4;
  __hip_atomic_fetch_add(np + 0, fs.x, __ATOMIC_RELAXED, __HIP_MEMORY_SCOPE_AGENT);
  __hip_atomic_fetch_add(np + 1, fs.y, __ATOMIC_RELAXED, __HIP_MEMORY_SCOPE_AGENT);
  __hip_atomic_fetch_add(np + 2, fs.z, __ATOMIC_RELAXED, __HIP_MEMORY_SCOPE_AGENT);
  __hip_atomic_fetch_add(np + 3, fs.w, __ATOMIC_RELAXED, __HIP_MEMORY_SCOPE_AGENT);

  float p = fs.x * tanhf(fd.x) + fs.y * tanhf(fd.y) +
            fs.z * tanhf(fd.z) + fs.w * tanhf(fd.w);
  // wave32 reduction
  for (int off = 16; off > 0; off >>= 1) p += __shfl_xor(p, off, 32);

  if (lane == 0) {
    e_raw[edge] = p;
    atomicMax(&mkey[d], f2key(p));
  }
}

__global__ void __launch_bounds__(256) k_exp(const float* __restrict__ e_raw,
                                             const int* __restrict__ dst,
                                             const unsigned* __restrict__ mkey,
                                             float* __restrict__ e_exp,
                                             float* __restrict__ denom) {
  int e = blockIdx.x * 256 + threadIdx.x;
  if (e >= N_EDGES) return;
  int d = dst[e];
  float v = __expf(e_raw[e] - key2f(mkey[d]));
  e_exp[e] = v;
  __hip_atomic_fetch_add(&denom[d], v, __ATOMIC_RELAXED, __HIP_MEMORY_SCOPE_AGENT);
}

__global__ void __launch_bounds__(256) k_norm(const float* __restrict__ e_exp,
                                              const int* __restrict__ dst,
                                              const float* __restrict__ denom,
                                              float* __restrict__ e_soft) {
  int e = blockIdx.x * 256 + threadIdx.x;
  if (e >= N_EDGES) return;
  e_soft[e] = e_exp[e] / denom[dst[e]];
}

// rst = (feat + neigh) @ W + b, exact fp32 via V_WMMA_F32_16X16X4_F32.
// Block: 256 threads = 8 waves. One 16-row M-tile per block; wave w owns the
// 16-column N-tile starting at w*16. A tile staged in LDS with stride 132
// (row*132 mod 64banks = row*4 -> conflict-free fragment reads).
__global__ void __launch_bounds__(256) k_gemm(const float* __restrict__ feat,
                                              const float* __restrict__ neigh,
                                              const float* __restrict__ Wm,
                                              const float* __restrict__ bias,
                                              float* __restrict__ rst) {
  __shared__ float Atile[16 * 132];

  const int m0 = blockIdx.x * 16;

  // Cooperative stage: A = feat + neigh (16 x 128 fp32)
#pragma unroll
  for (int j = 0; j < 8; ++j) {
    int i   = threadIdx.x + 256 * j;       // 0..2047
    int row = i >> 7;
    int col = i & 127;
    size_t g = (size_t)(m0 + row) * D_FEAT + col;
    Atile[row * 132 + col] = feat[g] + neigh[g];
  }
  __syncthreads();

  const int wave  = threadIdx.x >> 5;
  const int lane  = threadIdx.x & 31;
  const int n0    = wave * 16;
  const int mrow  = lane & 15;          // A: row; B: column (ISA 7.12.2 layout)
  const int kpair = (lane >> 4) * 2;    // lanes 16-31 hold K=+2,+3

  // C init = bias broadcast down the column (all 8 acc VGPRs share one column)
  v8f c;
  const float bv = bias[n0 + mrow];
#pragma unroll
  for (int v = 0; v < 8; ++v) c[v] = bv;

#pragma unroll 4
  for (int k = 0; k < D_FEAT; k += 4) {
    v2f a;
    const float* ap = &Atile[mrow * 132 + k + kpair];
    a.x = ap[0];                                   // K = k+kpair   (ds_load_b64)
    a.y = ap[1];                                   // K = k+kpair+1

    v2f bf;
    bf.x = Wm[(size_t)(k + kpair)     * D_FEAT + n0 + mrow];
    bf.y = Wm[(size_t)(k + kpair + 1) * D_FEAT + n0 + mrow];

    c = __builtin_amdgcn_wmma_f32_16x16x4_f32(false, a, false, bf,
                                              (short)0, c, false, false);
  }

  // D layout: VGPR v -> row v + 8*lane[4], col = lane&15
#pragma unroll
  for (int v = 0; v < 8; ++v) {
    int row = v + 8 * (lane >> 4);
    rst[(size_t)(m0 + row) * D_FEAT + n0 + (lane & 15)] = c[v];
  }
}

extern "C" void kernel_launch(void* const* d_in, const int* in_sizes, int n_in,
                              void* d_out, int out_size, void* d_ws, size_t ws_size,
                              hipStream_t stream) {
  (void)in_sizes; (void)n_in; (void)out_size; (void)ws_size;

  const float* feat = (const float*)d_in[0];
  const float* Wm   = (const float*)d_in[1];
  const float* bias = (const float*)d_in[2];
  const int*   src  = (const int*)d_in[3];
  const int*   dst  = (const int*)d_in[4];

  float* rst    = (float*)d_out;                               // [N, D]
  float* e_soft = (float*)d_out + (size_t)N_NODES * D_FEAT;    // [E]

  float*    ws    = (float*)d_ws;
  float*    neigh = ws;
  float*    e_raw = ws + (size_t)N_NODES * D_FEAT;
  float*    e_exp = e_raw + N_EDGES;
  unsigned* mkey  = (unsigned*)(e_exp + N_EDGES);
  float*    denom = (float*)(mkey + N_NODES);

  k_init<<<(N_NODES * D_FEAT + 255) / 256, 256, 0, stream>>>(neigh, mkey, denom);
  k_edge<<<N_EDGES / 8, 256, 0, stream>>>(feat, src, dst, neigh, e_raw, mkey);
  k_exp <<<(N_EDGES + 255) / 256, 256, 0, stream>>>(e_raw, dst, mkey, e_exp, denom);
  k_norm<<<(N_EDGES + 255) / 256, 256, 0, stream>>>(e_exp, dst, denom, e_soft);
  k_gemm<<<N_NODES / 16, 256, 0, stream>>>(feat, neigh, Wm, bias, rst);
}